// CortexNetwork_22058952032334
// MI455X (gfx1250) — compile-verified
//
#include <hip/hip_runtime.h>
#include <hip/hip_bf16.h>
#include <stdint.h>

// ---------------------------------------------------------------------------
// CortexNetwork on MI455X (gfx1250, wave32).
// Bandwidth-bound (~263 MB streamed once, ~11us HBM floor @ 23.3 TB/s).
// - TDM tensor_load_to_lds stages the gathered 16x24x24 image patch into LDS.
// - All reductions run through v_wmma_f32_16x16x4_f32 with B = ones
//   (D = A*1 + C accumulates row sums), weights streamed with NT b64 loads.
// ---------------------------------------------------------------------------

typedef float  v2f  __attribute__((ext_vector_type(2)));
typedef float  v8f  __attribute__((ext_vector_type(8)));
typedef unsigned int v4u __attribute__((ext_vector_type(4)));
typedef int    v8i  __attribute__((ext_vector_type(8)));
typedef int    v4i  __attribute__((ext_vector_type(4)));

#define C_    16
#define GX_   36
#define GY_   36
#define RF_   24
#define IMG_  64
#define NGRID (GX_ * GY_)          // 1296
#define PATCH (RF_ * RF_)          // 576
#define LAT   (GX_ * GY_)          // 1296 lateral elements per (c,i,j)

__device__ __forceinline__ v8f wmma_acc_ones(v2f a, v8f acc) {
  const v2f bones = {1.0f, 1.0f};
  // D = A(16x4) x B(4x16 = ones) + C  -> row sums replicated across columns
  return __builtin_amdgcn_wmma_f32_16x16x4_f32(false, a, false, bones,
                                               (short)0, acc, false, false);
}

__device__ __forceinline__ float wmma_fold(v8f acc) {
  float s = acc[0] + acc[1] + acc[2] + acc[3] + acc[4] + acc[5] + acc[6] + acc[7];
  // lanes 0-15 hold rows 0-7 partials (col = lane), lanes 16-31 rows 8-15.
  s += __shfl_xor(s, 16, 32);
  return s;  // every lane now holds the full sum
}

__global__ __launch_bounds__(256)
void cortex_kernel(const float* __restrict__ x,
                   const float* __restrict__ prev,
                   const float* __restrict__ wa,
                   const float* __restrict__ we,
                   const float* __restrict__ wi,
                   const int*   __restrict__ rx,
                   const int*   __restrict__ ry,
                   float* __restrict__ out)
{
  __shared__ float s_patch[C_ * PATCH];  // 36864 B, packed [c][u][v]
  __shared__ float s_res[48];

  const int bid = blockIdx.x;            // (i,j) cell
  const int i   = bid / GY_;
  const int j   = bid - i * GY_;
  const int tid = threadIdx.x;

  // ---- Stage gathered patch x[c, rx[i]+u, ry[j]+v] into LDS via TDM -------
  if (tid == 0) {
    const int rxi = rx[i];
    const int ryj = ry[j];
    const uint64_t gaddr =
        (uint64_t)(uintptr_t)(x + (size_t)rxi * IMG_ + (size_t)ryj);
    const unsigned ldsOff = (unsigned)(uintptr_t)(&s_patch[0]);

    v4u g0;
    g0[0] = 1u;                                    // count=1, user descriptor
    g0[1] = ldsOff;                                // lds_addr
    g0[2] = (unsigned)(gaddr & 0xffffffffull);     // global_addr[31:0]
    g0[3] = (unsigned)((gaddr >> 32) & 0x01ffffffull) | 0x80000000u; // type=2

    v8i g1;
    g1[0] = (int)(2u << 16);                       // data_size = 4 bytes
    g1[1] = (int)((IMG_ & 0xffff) << 16);          // tensor_dim0 lo16
    g1[2] = (int)(((unsigned)IMG_ >> 16) | ((IMG_ & 0xffff) << 16)); // d0 hi | d1 lo
    g1[3] = (int)(((unsigned)IMG_ >> 16) | ((unsigned)RF_ << 16));   // d1 hi | tile_dim0
    g1[4] = (int)((unsigned)RF_ | ((unsigned)C_ << 16));             // tile_dim1 | tile_dim2
    g1[5] = IMG_;                                  // tensor_dim0_stride = 64
    g1[6] = (int)((IMG_ * IMG_) << 16);            // d0s hi16 | tensor_dim1_stride lo16 (4096)
    g1[7] = (int)((unsigned)(IMG_ * IMG_) >> 16);  // tensor_dim1_stride[47:16]

    v4i g2;
    g2[0] = C_;                                    // tensor_dim2 = 16 channels
    g2[1] = 0; g2[2] = 0; g2[3] = 0;               // no dim3 / no iterate
    v4i g3 = {0, 0, 0, 0};

#if __has_include(<hip/amd_detail/amd_gfx1250_TDM.h>)
    v8i g4 = {0, 0, 0, 0, 0, 0, 0, 0};
    __builtin_amdgcn_tensor_load_to_lds(g0, g1, g2, g3, g4, 0);
#else
    __builtin_amdgcn_tensor_load_to_lds(g0, g1, g2, g3, 0);
#endif
    __builtin_amdgcn_s_wait_tensorcnt(0);
  }
  __syncthreads();

  // ---- 48 reduction tasks (16 channels x {aff, exc-rowsum, inh-rowsum}) ---
  const int wave = tid >> 5;
  const int lane = tid & 31;
  const int row  = lane & 15;
  const int sub  = lane >> 4;
  const int off0 = row * 4 + sub * 2;   // A[m,k]: m=row, k = {0,1} or {2,3}
  const v2f vzero = {0.0f, 0.0f};

  for (int t = wave; t < 48; t += 8) {  // wave-uniform: EXEC stays all-ones
    const int c = t / 3;
    const int m = t - 3 * c;
    v8f acc = {0.f, 0.f, 0.f, 0.f, 0.f, 0.f, 0.f, 0.f};

    if (m == 0) {
      // afferent: dot(patch, weights) over 576 elements (exactly 9 chunks)
      const float* gb = wa + (((size_t)c * GX_ + i) * GY_ + j) * PATCH;
      const float* lb = &s_patch[c * PATCH];
      __builtin_prefetch(gb, 0, 1);
#pragma unroll
      for (int ch = 0; ch < 9; ++ch) {
        const int e = ch * 64 + off0;
        v2f wv = __builtin_nontemporal_load((const v2f*)(gb + e));
        v2f pv = *(const v2f*)(lb + e);
        acc = wmma_acc_ones(wv * pv, acc);
      }
    } else {
      // lateral row sum over 1296 contiguous floats (21 chunks, last masked)
      const float* gb = (m == 1 ? we : wi)
                      + (((size_t)c * GX_ + i) * GY_ + j) * (size_t)LAT;
      __builtin_prefetch(gb, 0, 1);
#pragma unroll
      for (int ch = 0; ch < 21; ++ch) {
        const int e  = ch * 64 + off0;
        const int eS = (e <= LAT - 2) ? e : (LAT - 2);     // clamp address
        v2f a = __builtin_nontemporal_load((const v2f*)(gb + eS));
        a = (e > LAT - 2) ? vzero : a;                     // mask, not EXEC
        acc = wmma_acc_ones(a, acc);
      }
    }

    const float s = wmma_fold(acc);
    if (lane == 0) s_res[t] = s;
  }
  __syncthreads();

  // ---- Combine + relu + broadcast over channels ---------------------------
  if (tid < C_) {
    float aff = 0.f, exc = 0.f, inh = 0.f;
#pragma unroll
    for (int c = 0; c < C_; ++c) {
      const float p = prev[(size_t)c * NGRID + bid];
      aff += s_res[c * 3 + 0];
      exc += p * s_res[c * 3 + 1];
      inh += p * s_res[c * 3 + 2];
    }
    const float act = fmaxf(aff + 0.9f * exc - 0.9f * inh, 0.0f);
    out[(size_t)tid * NGRID + bid] = act;
  }
}

extern "C" void kernel_launch(void* const* d_in, const int* in_sizes, int n_in,
                              void* d_out, int out_size, void* d_ws, size_t ws_size,
                              hipStream_t stream) {
  (void)in_sizes; (void)n_in; (void)d_ws; (void)ws_size; (void)out_size;
  const float* x    = (const float*)d_in[0];
  const float* prev = (const float*)d_in[1];
  const float* wa   = (const float*)d_in[2];
  const float* we   = (const float*)d_in[3];
  const float* wi   = (const float*)d_in[4];
  const int*   rx   = (const int*)d_in[5];
  const int*   ry   = (const int*)d_in[6];
  float* out = (float*)d_out;

  cortex_kernel<<<dim3(NGRID), dim3(256), 0, stream>>>(x, prev, wa, we, wi,
                                                       rx, ry, out);
}